// VectorQuantizerEMA_38371237822826
// MI455X (gfx1250) — compile-verified
//
#include <hip/hip_runtime.h>

typedef __attribute__((ext_vector_type(2))) float v2f;
typedef __attribute__((ext_vector_type(8))) float v8f;

#define K_CODES 64
#define DIM     128
#define NTOK    16384   // 4 * 4096 tokens
#define TPB     4096    // tokens per batch (boundary masking for pairs/triples)

// ---- workspace layout (float offsets), total ~9.5 MB ----
#define WS_Q      0u          // 16384*64  log-softmax q
#define WS_DIST   1048576u    // 16384*64  shifted distances (esq - 2*dot)
#define WS_C      2097152u    // 64*64*64  third-order tensor C
#define WS_CPAIR  2359296u    // 64*64     pairwise C_pair
#define WS_ESQ    2363392u    // 64        ||e_c||^2
#define WS_CNT    2363456u    // 64        code usage counts (float)
#define WS_MSE    2363520u    // scalar accumulators
#define WS_CPSUM  2363521u
#define WS_H2     2363522u

static __device__ __forceinline__ v8f wmma_f32(v2f a, v2f b, v8f c) {
  // V_WMMA_F32_16X16X4_F32: D = A(16x4) * B(4x16) + C(16x16), fp32
  return __builtin_amdgcn_wmma_f32_16x16x4_f32(false, a, false, b, (short)0, c,
                                               false, false);
}

// ---------------- Tensor Data Mover (TDM) support ----------------
#if defined(__has_builtin)
#if __has_builtin(__builtin_amdgcn_tensor_load_to_lds) && \
    __has_builtin(__builtin_amdgcn_s_wait_tensorcnt)
#define USE_TDM 1
#endif
#endif
#ifndef USE_TDM
#define USE_TDM 0
#endif

#if USE_TDM
typedef __attribute__((ext_vector_type(4))) unsigned int tdm_v4u;
typedef __attribute__((ext_vector_type(8))) int          tdm_v8i;
typedef __attribute__((ext_vector_type(4))) int          tdm_v4i;

// 2D TDM load: tile_rows x 64 f32 rows from gsrc (row stride 64) into LDS.
// tensor_rows bounds dim1 so OOB rows read as zero (ISA 8.2).
static __device__ __forceinline__ void tdm_load_rows(const float* gsrc, float* ldst,
                                                     int tile_rows, int tensor_rows) {
  // Flat LDS aperture address: low 32 bits are the LDS byte address (ISA 10.2).
  unsigned lds_off = (unsigned)(size_t)ldst;
  unsigned long long ga = (unsigned long long)(size_t)gsrc;
  tdm_v4u g0;
  g0[0] = 1u;                                                // count=1 valid user D#
  g0[1] = lds_off;                                           // lds_addr
  g0[2] = (unsigned)ga;                                      // global_addr[31:0]
  g0[3] = (unsigned)((ga >> 32) & 0x01FFFFFFu) | (2u << 30); // addr[56:32] | type=2
  tdm_v8i g1;
  g1[0] = (int)(2u << 16);                                   // data_size=4B; no multicast
  g1[1] = (int)(64u << 16);                                  // tensor_dim0 = 64 cols
  g1[2] = (int)(((unsigned)tensor_rows & 0xFFFFu) << 16);    // tensor_dim1 lo16
  g1[3] = (int)((((unsigned)tensor_rows >> 16) & 0xFFFFu)    // tensor_dim1 hi16
                | (64u << 16));                              // tile_dim0 = 64
  g1[4] = (int)((unsigned)tile_rows & 0xFFFFu);              // tile_dim1; tile_dim2=0
  g1[5] = 64;                                                // tensor_dim0_stride = 64
  g1[6] = 0;                                                 // stride hi / dim1_stride
  g1[7] = 0;
  tdm_v4i gz4 = {0, 0, 0, 0};                                // 2D tensor: groups 2/3 unused
  tdm_v8i gz8 = {0, 0, 0, 0, 0, 0, 0, 0};
  // 6-arg toolchain form: (g0, g1, g2, g3, extra_group, cpol)
  __builtin_amdgcn_tensor_load_to_lds(g0, g1, gz4, gz4, gz8, 0);
}
#endif

// ---- init: embedding norms + zero accumulators (64 threads) ----
__global__ void init_kernel(const float* __restrict__ emb, float* __restrict__ ws) {
  int k = threadIdx.x;
  float s = 0.f;
  for (int d = 0; d < DIM; ++d) { float e = emb[k * DIM + d]; s += e * e; }
  ws[WS_ESQ + k] = s;
  ws[WS_CNT + k] = 0.f;
  if (k < 3) ws[WS_MSE + k] = 0.f;   // MSE, CPSUM, H2
}

// ---- distance GEMM: dist'[t,c] = ||e_c||^2 - 2 * x_t . e_c  (WMMA fp32) ----
// grid (NTOK/16, 64/16), block 32 (one wave per 16x16 tile)
__global__ void dist_gemm(const float* __restrict__ x, const float* __restrict__ emb,
                          float* __restrict__ ws) {
  const int lane = threadIdx.x;
  const int tok0 = blockIdx.x * 16;
  const int c0   = blockIdx.y * 16;
  const int mn   = lane & 15;          // A row (token) / B col (code)
  const int kb   = (lane >> 4) * 2;    // K phase: lanes 0-15 -> K 0,1 ; 16-31 -> K 2,3
  const float* xrow = x   + (size_t)(tok0 + mn) * DIM;
  const float* erow = emb + (size_t)(c0   + mn) * DIM;
  v8f acc = {};
  for (int d = 0; d < DIM; d += 4) {
    v2f a, b;
    a.x = xrow[d + kb];  a.y = xrow[d + kb + 1];
    b.x = erow[d + kb];  b.y = erow[d + kb + 1];
    acc = wmma_f32(a, b, acc);
  }
  const int col = c0 + mn;
  const float esq = ws[WS_ESQ + col];
  const int hi = lane >> 4;
  for (int r = 0; r < 8; ++r) {
    int row = tok0 + r + 8 * hi;       // C/D layout: VGPR r -> M=r (lo half) / r+8 (hi half)
    ws[WS_DIST + (size_t)row * K_CODES + col] = esq - 2.0f * acc[r];
  }
}

// ---- per-token: log_softmax -> q, argmin -> quantized + counts + MSE ----
__global__ void token_reduce(const float* __restrict__ x, const float* __restrict__ emb,
                             float* __restrict__ ws, float* __restrict__ out) {
  const int lane = threadIdx.x & 31;
  const int t = blockIdx.x * 8 + (threadIdx.x >> 5);
  const float* dr = ws + WS_DIST + (size_t)t * K_CODES;
  float d0 = dr[lane], d1 = dr[lane + 32];

  // logsumexp over 64 codes (shift-invariant, so dist' is fine)
  float mx = fmaxf(d0, d1);
  for (int off = 16; off; off >>= 1) mx = fmaxf(mx, __shfl_xor(mx, off));
  float se = __expf(d0 - mx) + __expf(d1 - mx);
  for (int off = 16; off; off >>= 1) se += __shfl_xor(se, off);
  float lse = mx + __logf(se);
  float* qr = ws + WS_Q + (size_t)t * K_CODES;
  qr[lane]      = d0 - lse;
  qr[lane + 32] = d1 - lse;

  // argmin with lowest-index tie-break
  float mv; int mi;
  if (d0 <= d1) { mv = d0; mi = lane; } else { mv = d1; mi = lane + 32; }
  for (int off = 16; off; off >>= 1) {
    float ov = __shfl_xor(mv, off);
    int   oi = __shfl_xor(mi, off);
    if (ov < mv || (ov == mv && oi < mi)) { mv = ov; mi = oi; }
  }
  if (lane == 0) atomicAdd(ws + WS_CNT + mi, 1.0f);

  // quantized output + squared-error accumulation
  const float* ev = emb + (size_t)mi * DIM;
  const float* xv = x   + (size_t)t  * DIM;
  float lsum = 0.f;
  for (int j = lane * 4; j < lane * 4 + 4; ++j) {
    float e = ev[j];
    float df = e - xv[j];
    out[(size_t)t * DIM + j] = e;
    lsum += df * df;
  }
  for (int off = 16; off; off >>= 1) lsum += __shfl_xor(lsum, off);
  if (lane == 0) atomicAdd(ws + WS_MSE, lsum);
}

// ---- C_pair[k,n] = sum_t q[t,k] q[t+1,n] (batch-boundary masked), WMMA fp32 ----
__global__ void cpair_gemm(float* __restrict__ ws) {
  const int lane = threadIdx.x;
  const int m0 = blockIdx.x * 16, n0 = blockIdx.y * 16;
  const int mn = lane & 15, kb = (lane >> 4) * 2;
  const float* q = ws + WS_Q;
  v8f acc = {};
  for (int t = 0; t < NTOK; t += 4) {
    int t0 = t + kb, t1 = t0 + 1;
    bool v0 = ((t0 & (TPB - 1)) != TPB - 1);
    bool v1 = ((t1 & (TPB - 1)) != TPB - 1);
    v2f a, b;
    a.x = v0 ? q[(size_t)t0 * 64 + m0 + mn] : 0.f;
    a.y = v1 ? q[(size_t)t1 * 64 + m0 + mn] : 0.f;
    b.x = v0 ? q[(size_t)(t0 + 1) * 64 + n0 + mn] : 0.f;
    b.y = v1 ? q[(size_t)(t1 + 1) * 64 + n0 + mn] : 0.f;
    acc = wmma_f32(a, b, acc);
  }
  const int hi = lane >> 4;
  float lsum = 0.f;
  for (int r = 0; r < 8; ++r) {
    int row = m0 + r + 8 * hi;
    ws[WS_CPAIR + row * 64 + n0 + mn] = acc[r];
    lsum += acc[r];
  }
  atomicAdd(ws + WS_CPSUM, lsum);
}

// ---- C[k,m,n] = sum_t q[t,k] q[t+1,m] q[t+2,n]  (the 8.6 GFLOP core) ----
// TDM double-buffered LDS staging: block = 8 waves sharing one q chunk,
// wave w computes k = blockIdx.z*8 + w on the (m0,n0) 16x16 tile.
#define CH      128            // chunk rows per TDM transfer
#define CHROWS  (CH + 2)       // +2 halo rows so t+1 / t+2 stay in-buffer
#define NCH     (NTOK / CH)

__global__ void __launch_bounds__(256) ctriple_gemm(float* __restrict__ ws) {
  __shared__ float lds[2][CHROWS * 64];   // 2 x 32.5 KB double buffer
  const int tid  = threadIdx.x;
  const int lane = tid & 31;
  const int wave = tid >> 5;
  const int m0 = blockIdx.x * 16, n0 = blockIdx.y * 16;
  const int k  = blockIdx.z * 8 + wave;
  const int mn = lane & 15, kb = (lane >> 4) * 2;
  const float* q = ws + WS_Q;

  // prologue: chunk 0 into buffer 0
#if USE_TDM
  if (tid < 32) {                         // one wave issues the DMA (TDM ignores EXEC)
    tdm_load_rows(q, &lds[0][0], CHROWS, NTOK);
    __builtin_amdgcn_s_wait_tensorcnt(0);
  }
#else
  for (int i = tid; i < CHROWS * 64; i += 256) lds[0][i] = q[i];
#endif
  __syncthreads();

  v8f acc = {};
  for (int c = 0; c < NCH; ++c) {
    const int base = c * CH;
    const float* buf = lds[c & 1];
    if (c + 1 < NCH) {                    // kick off next chunk; overlaps with WMMA below
#if USE_TDM
      if (tid < 32)
        tdm_load_rows(q + (size_t)(base + CH) * 64, &lds[(c + 1) & 1][0],
                      CHROWS, NTOK - (base + CH));
#else
      const float* src = q + (size_t)(base + CH) * 64;
      int lim = (NTOK - (base + CH)) * 64;
      if (lim > CHROWS * 64) lim = CHROWS * 64;
      for (int i = tid; i < lim; i += 256) lds[(c + 1) & 1][i] = src[i];
#endif
    }
    for (int tt = 0; tt < CH; tt += 4) {
      int r0 = tt + kb, r1 = r0 + 1;
      int t0 = base + r0, t1 = t0 + 1;
      bool v0 = ((t0 & (TPB - 1)) <= TPB - 3);   // triple fits inside one batch
      bool v1 = ((t1 & (TPB - 1)) <= TPB - 3);
      float w0 = buf[r0 * 64 + k];               // q0 weight (broadcast read)
      float w1 = buf[r1 * 64 + k];
      v2f a, b;
      a.x = v0 ? w0 * buf[(r0 + 1) * 64 + m0 + mn] : 0.f;   // q1 column slice
      a.y = v1 ? w1 * buf[(r1 + 1) * 64 + m0 + mn] : 0.f;
      b.x = v0 ? buf[(r0 + 2) * 64 + n0 + mn] : 0.f;        // q2 column slice
      b.y = v1 ? buf[(r1 + 2) * 64 + n0 + mn] : 0.f;
      acc = wmma_f32(a, b, acc);
    }
#if USE_TDM
    if (tid < 32) __builtin_amdgcn_s_wait_tensorcnt(0);
#endif
    __syncthreads();                     // publish next buffer / retire this one
  }
  const int hi = lane >> 4;
  for (int r = 0; r < 8; ++r) {
    int row = m0 + r + 8 * hi;
    ws[WS_C + ((size_t)k * 64 + row) * 64 + n0 + mn] = acc[r];
  }
}

// ---- entropy over n per (k,m) row + pi_hat weighting ----
__global__ void entropy_kernel(float* __restrict__ ws) {
  const int k = blockIdx.x, m = threadIdx.x;
  const float* crow = ws + WS_C + ((size_t)k * 64 + m) * 64;
  float s = 0.f;
  for (int n = 0; n < 64; ++n) s += crow[n] + 0.001f;
  float inv = 1.0f / (s + 1e-8f);
  float h = 0.f;
  for (int n = 0; n < 64; ++n) {
    float T = (crow[n] + 0.001f) * inv;
    h -= T * __logf(T + 1e-8f);
  }
  float pi = ws[WS_CPAIR + k * 64 + m] / (ws[WS_CPSUM] + 1e-8f);
  atomicAdd(ws + WS_H2, pi * h);
}

// ---- total_loss = 1.25*mse + KL + 0.1*H2  -> out[NTOK*DIM] ----
__global__ void finalize_kernel(const float* __restrict__ prior, float* __restrict__ ws,
                                float* __restrict__ out) {
  float kl = 0.f;
  for (int k = 0; k < 64; ++k) {
    float p = ws[WS_CNT + k] * (1.0f / (float)NTOK);
    kl += p * (__logf(p + 1e-10f) - __logf(prior[k] + 1e-10f));
  }
  float mse = ws[WS_MSE] * (1.0f / (float)(NTOK * DIM));
  out[(size_t)NTOK * DIM] = 1.25f * mse + 1.0f * kl + 0.1f * ws[WS_H2];
}

extern "C" void kernel_launch(void* const* d_in, const int* in_sizes, int n_in,
                              void* d_out, int out_size, void* d_ws, size_t ws_size,
                              hipStream_t stream) {
  const float* x     = (const float*)d_in[0];   // (4,4096,128)
  // d_in[1] = batch (unused by the math)
  const float* emb   = (const float*)d_in[2];   // (64,128)
  const float* prior = (const float*)d_in[3];   // (64,)
  float* ws  = (float*)d_ws;
  float* out = (float*)d_out;                   // [quantized(16384*128) | total_loss]

  init_kernel   <<<1, 64, 0, stream>>>(emb, ws);
  dist_gemm     <<<dim3(NTOK / 16, K_CODES / 16), 32, 0, stream>>>(x, emb, ws);
  token_reduce  <<<NTOK / 8, 256, 0, stream>>>(x, emb, ws, out);
  cpair_gemm    <<<dim3(4, 4), 32, 0, stream>>>(ws);
  ctriple_gemm  <<<dim3(4, 4, 8), 256, 0, stream>>>(ws);
  entropy_kernel<<<64, 64, 0, stream>>>(ws);
  finalize_kernel<<<1, 1, 0, stream>>>(prior, ws, out);
}